// EncoderDecoder_25958782337206
// MI455X (gfx1250) — compile-verified
//
#include <hip/hip_runtime.h>
#include <hip/hip_bf16.h>
#include <math.h>

// ---------------------------------------------------------------------------
// GRU seq2seq encoder-decoder with attention, batch=1, for MI455X (gfx1250).
//
// Strategy (see analysis):
//  * Batch all time-independent matvecs into WMMA f32 GEMMs:
//      encGI  = enc_x  @ enc_Wi^T + enc_bi      [256 x 3072]
//      attn_e = dec_e  @ attn_W[:, :H]^T + b    [256 x 256]
//      logits = H2     @ out_W^T + out_b        [256 x 32000]  (the big one)
//  * Sequential recurrences run as one small grid-wide kernel per step;
//    the 12.6MB recurrent weight matrices stay resident in the 192MB L2.
//  * Final log-softmax / NLL reduction over the materialized logits.
// Workspace: ~38.5 MB.
// ---------------------------------------------------------------------------

#define HD   1024
#define HD3  3072
#define LL   256
#define VOUT 32000
#define START_ID 1

typedef float v2f __attribute__((ext_vector_type(2)));
typedef float v8f __attribute__((ext_vector_type(8)));

__device__ __forceinline__ float wave_sum32(float v) {
  #pragma unroll
  for (int off = 16; off > 0; off >>= 1) v += __shfl_xor(v, off, 32);
  return v;
}
__device__ __forceinline__ float sigm(float x) { return 1.0f / (1.0f + __expf(-x)); }

// ---------------------------------------------------------------------------
// Generic WMMA f32 GEMM:  C[M x N] = A[M x K] * B^T  (B is N x K row-major)
// Optional bias[n] added per column. Tiles: wave = 16M x 64N, block(128) =
// 16M x 256N. Requires M%16==0, N%256==0, K%4==0, lda/ldb even.
// A-frag (16x4 f32, ISA 7.12.2): lane m=lane&15, half=lane>>4 selects K pair.
// C-frag: acc[r] -> M = r + 8*half, N = lane&15.
// ---------------------------------------------------------------------------
__global__ void wmma_gemm_nt(const float* __restrict__ A, int lda,
                             const float* __restrict__ B, int ldb,
                             const float* __restrict__ bias,
                             float* __restrict__ C, int ldc, int K) {
  const int wave = threadIdx.x >> 5;
  const int lane = threadIdx.x & 31;
  const int half = lane >> 4;
  const int r16  = lane & 15;
  const int m0 = blockIdx.y * 16;
  const int n0 = blockIdx.x * 256 + wave * 64;

  const float* Arow = A + (size_t)(m0 + r16) * lda;
  const float* B0 = B + (size_t)(n0 +  0 + r16) * ldb;
  const float* B1 = B + (size_t)(n0 + 16 + r16) * ldb;
  const float* B2 = B + (size_t)(n0 + 32 + r16) * ldb;
  const float* B3 = B + (size_t)(n0 + 48 + r16) * ldb;

  v8f acc[4] = {};
  for (int k0 = 0; k0 < K; k0 += 4) {
    const int kb = k0 + half * 2;
    v2f a  = *(const v2f*)(Arow + kb);
    v2f b0 = *(const v2f*)(B0 + kb);
    v2f b1 = *(const v2f*)(B1 + kb);
    v2f b2 = *(const v2f*)(B2 + kb);
    v2f b3 = *(const v2f*)(B3 + kb);
    acc[0] = __builtin_amdgcn_wmma_f32_16x16x4_f32(false, a, false, b0, (short)0, acc[0], false, false);
    acc[1] = __builtin_amdgcn_wmma_f32_16x16x4_f32(false, a, false, b1, (short)0, acc[1], false, false);
    acc[2] = __builtin_amdgcn_wmma_f32_16x16x4_f32(false, a, false, b2, (short)0, acc[2], false, false);
    acc[3] = __builtin_amdgcn_wmma_f32_16x16x4_f32(false, a, false, b3, (short)0, acc[3], false, false);
  }
  const int mrow = m0 + half * 8;
  #pragma unroll
  for (int j = 0; j < 4; ++j) {
    const int n = n0 + j * 16 + r16;
    const float bv = bias ? bias[n] : 0.0f;
    #pragma unroll
    for (int r = 0; r < 8; ++r)
      C[(size_t)(mrow + r) * ldc + n] = acc[j][r] + bv;
  }
}

// ---------------------------------------------------------------------------
// Init: gather embeddings for encoder inputs & decoder (teacher-forced)
// inputs; zero the initial hidden state.
// ---------------------------------------------------------------------------
__global__ void init_gather(const int* __restrict__ input_ids,
                            const int* __restrict__ target_ids,
                            const float* __restrict__ enc_emb,
                            const float* __restrict__ dec_emb,
                            float* __restrict__ enc_x,
                            float* __restrict__ dec_e,
                            float* __restrict__ h0) {
  const int t = blockIdx.x;
  const int enc_tok = input_ids[t];
  const int dec_tok = (t == 0) ? START_ID : target_ids[t - 1];
  for (int j = threadIdx.x; j < HD; j += blockDim.x) {
    enc_x[(size_t)t * HD + j] = enc_emb[(size_t)enc_tok * HD + j];
    dec_e[(size_t)t * HD + j] = dec_emb[(size_t)dec_tok * HD + j];
    if (t == 0) h0[j] = 0.0f;
  }
}

// ---------------------------------------------------------------------------
// One GRU step with precomputed input gates gi (3H). One wave per output j.
// gh = Wh @ h + bh computed as 3 dots per j.
// ---------------------------------------------------------------------------
__global__ void gru_step_pre(const float* __restrict__ Wh,
                             const float* __restrict__ bh,
                             const float* __restrict__ gi,    // 3H for this step
                             const float* __restrict__ hprev,
                             float* __restrict__ hout) {
  const int lane = threadIdx.x & 31;
  const int j = blockIdx.x * (blockDim.x >> 5) + (threadIdx.x >> 5);
  const float* w0 = Wh + (size_t)j * HD;
  const float* w1 = Wh + (size_t)(j + HD) * HD;
  const float* w2 = Wh + (size_t)(j + 2 * HD) * HD;
  float s0 = 0.f, s1 = 0.f, s2 = 0.f;
  for (int k = lane; k < HD; k += 32) {
    const float hv = hprev[k];
    s0 = fmaf(w0[k], hv, s0);
    s1 = fmaf(w1[k], hv, s1);
    s2 = fmaf(w2[k], hv, s2);
  }
  s0 = wave_sum32(s0); s1 = wave_sum32(s1); s2 = wave_sum32(s2);
  if (lane == 0) {
    const float r = sigm(gi[j]          + s0 + bh[j]);
    const float z = sigm(gi[j + HD]     + s1 + bh[j + HD]);
    const float n = tanhf(gi[j + 2*HD]  + r * (s2 + bh[j + 2*HD]) );
    const float h = hprev[j];
    hout[j] = (1.0f - z) * n + z * h;
  }
}

// ---------------------------------------------------------------------------
// Decoder GRU step: input gates computed on the fly from c. 6 dots per j.
// ---------------------------------------------------------------------------
__global__ void gru_step_full(const float* __restrict__ Wi,
                              const float* __restrict__ Wh,
                              const float* __restrict__ bi,
                              const float* __restrict__ bh,
                              const float* __restrict__ c,
                              const float* __restrict__ hprev,
                              float* __restrict__ hout) {
  const int lane = threadIdx.x & 31;
  const int j = blockIdx.x * (blockDim.x >> 5) + (threadIdx.x >> 5);
  const float* wi0 = Wi + (size_t)j * HD;
  const float* wi1 = Wi + (size_t)(j + HD) * HD;
  const float* wi2 = Wi + (size_t)(j + 2 * HD) * HD;
  const float* wh0 = Wh + (size_t)j * HD;
  const float* wh1 = Wh + (size_t)(j + HD) * HD;
  const float* wh2 = Wh + (size_t)(j + 2 * HD) * HD;
  float i0 = 0.f, i1 = 0.f, i2 = 0.f, g0 = 0.f, g1 = 0.f, g2 = 0.f;
  for (int k = lane; k < HD; k += 32) {
    const float cv = c[k], hv = hprev[k];
    i0 = fmaf(wi0[k], cv, i0);
    i1 = fmaf(wi1[k], cv, i1);
    i2 = fmaf(wi2[k], cv, i2);
    g0 = fmaf(wh0[k], hv, g0);
    g1 = fmaf(wh1[k], hv, g1);
    g2 = fmaf(wh2[k], hv, g2);
  }
  i0 = wave_sum32(i0); i1 = wave_sum32(i1); i2 = wave_sum32(i2);
  g0 = wave_sum32(g0); g1 = wave_sum32(g1); g2 = wave_sum32(g2);
  if (lane == 0) {
    const float r = sigm(i0 + bi[j]        + g0 + bh[j]);
    const float z = sigm(i1 + bi[j + HD]   + g1 + bh[j + HD]);
    const float n = tanhf(i2 + bi[j + 2*HD] + r * (g2 + bh[j + 2*HD]));
    const float h = hprev[j];
    hout[j] = (1.0f - z) * n + z * h;
  }
}

// ---------------------------------------------------------------------------
// Attention scores: scores[l] = attn_e_row[l] + attn_W[l, H:2H] . hprev
// One wave per l (L=256 waves total).
// ---------------------------------------------------------------------------
__global__ void attn_score(const float* __restrict__ attn_W,
                           const float* __restrict__ attn_e_row,  // L
                           const float* __restrict__ hprev,
                           float* __restrict__ scores) {
  const int lane = threadIdx.x & 31;
  const int l = blockIdx.x * (blockDim.x >> 5) + (threadIdx.x >> 5);
  const float* w = attn_W + (size_t)l * (2 * HD) + HD;
  float s = 0.f;
  for (int k = lane; k < HD; k += 32) s = fmaf(w[k], hprev[k], s);
  s = wave_sum32(s);
  if (lane == 0) scores[l] = attn_e_row[l] + s;
}

// ---------------------------------------------------------------------------
// Softmax over L scores + context ctx = a @ enc_outs. Single block, 256 thr.
// ---------------------------------------------------------------------------
__global__ void softmax_ctx(const float* __restrict__ scores,
                            const float* __restrict__ enc_outs,
                            float* __restrict__ ctx) {
  __shared__ float red[LL];
  __shared__ float aa[LL];
  const int tid = threadIdx.x;
  const float s = scores[tid];
  red[tid] = s;
  __syncthreads();
  for (int off = LL / 2; off > 0; off >>= 1) {
    if (tid < off) red[tid] = fmaxf(red[tid], red[tid + off]);
    __syncthreads();
  }
  const float mx = red[0];
  __syncthreads();
  const float e = __expf(s - mx);
  red[tid] = e;
  __syncthreads();
  for (int off = LL / 2; off > 0; off >>= 1) {
    if (tid < off) red[tid] += red[tid + off];
    __syncthreads();
  }
  aa[tid] = e / red[0];
  __syncthreads();
  for (int j = tid; j < HD; j += LL) {
    float acc = 0.f;
    for (int l = 0; l < LL; ++l) acc = fmaf(aa[l], enc_outs[(size_t)l * HD + j], acc);
    ctx[j] = acc;
  }
}

// ---------------------------------------------------------------------------
// Combine: c[j] = relu(comb_W[j,:H].e + comb_W[j,H:].ctx + comb_b[j])
// One wave per j.
// ---------------------------------------------------------------------------
__global__ void combine(const float* __restrict__ comb_W,
                        const float* __restrict__ comb_b,
                        const float* __restrict__ e,
                        const float* __restrict__ ctx,
                        float* __restrict__ c) {
  const int lane = threadIdx.x & 31;
  const int j = blockIdx.x * (blockDim.x >> 5) + (threadIdx.x >> 5);
  const float* w = comb_W + (size_t)j * (2 * HD);
  float s = 0.f;
  for (int k = lane; k < HD; k += 32) {
    s = fmaf(w[k], e[k], s);
    s = fmaf(w[HD + k], ctx[k], s);
  }
  s = wave_sum32(s);
  if (lane == 0) c[j] = fmaxf(s + comb_b[j], 0.0f);
}

// ---------------------------------------------------------------------------
// Per-step NLL from materialized logits: nll[t] = logsumexp(row) - row[tgt].
// One block per t.
// ---------------------------------------------------------------------------
__global__ void row_nll(const float* __restrict__ logits,
                        const int* __restrict__ target_ids,
                        float* __restrict__ nll) {
  __shared__ float red[256];
  const int t = blockIdx.x;
  const int tid = threadIdx.x;
  const float* row = logits + (size_t)t * VOUT;
  float m = -INFINITY;
  for (int v = tid; v < VOUT; v += 256) m = fmaxf(m, row[v]);
  red[tid] = m;
  __syncthreads();
  for (int off = 128; off > 0; off >>= 1) {
    if (tid < off) red[tid] = fmaxf(red[tid], red[tid + off]);
    __syncthreads();
  }
  const float mx = red[0];
  __syncthreads();
  float s = 0.f;
  for (int v = tid; v < VOUT; v += 256) s += __expf(row[v] - mx);
  red[tid] = s;
  __syncthreads();
  for (int off = 128; off > 0; off >>= 1) {
    if (tid < off) red[tid] += red[tid + off];
    __syncthreads();
  }
  if (tid == 0) {
    const float logZ = mx + logf(red[0]);
    nll[t] = logZ - row[target_ids[t]];
  }
}

__global__ void final_sum(const float* __restrict__ nll, float* __restrict__ out) {
  __shared__ float red[LL];
  const int tid = threadIdx.x;
  red[tid] = nll[tid];
  __syncthreads();
  for (int off = LL / 2; off > 0; off >>= 1) {
    if (tid < off) red[tid] += red[tid + off];
    __syncthreads();
  }
  if (tid == 0) out[0] = red[0];
}

// ---------------------------------------------------------------------------
extern "C" void kernel_launch(void* const* d_in, const int* in_sizes, int n_in,
                              void* d_out, int out_size, void* d_ws, size_t ws_size,
                              hipStream_t stream) {
  const int*   input_ids  = (const int*)  d_in[0];
  const int*   target_ids = (const int*)  d_in[1];
  const float* enc_emb    = (const float*)d_in[2];
  const float* enc_Wi     = (const float*)d_in[3];
  const float* enc_Wh     = (const float*)d_in[4];
  const float* enc_bi     = (const float*)d_in[5];
  const float* enc_bh     = (const float*)d_in[6];
  const float* dec_emb    = (const float*)d_in[7];
  const float* dec_Wi     = (const float*)d_in[8];
  const float* dec_Wh     = (const float*)d_in[9];
  const float* dec_bi     = (const float*)d_in[10];
  const float* dec_bh     = (const float*)d_in[11];
  const float* attn_W     = (const float*)d_in[12];
  const float* attn_b     = (const float*)d_in[13];
  const float* comb_W     = (const float*)d_in[14];
  const float* comb_b     = (const float*)d_in[15];
  const float* out_W      = (const float*)d_in[16];
  const float* out_b      = (const float*)d_in[17];
  float* out = (float*)d_out;

  // Workspace carve-up (floats); total ~38.5 MB.
  float* f        = (float*)d_ws;
  float* enc_x    = f;                         // L*H
  float* encGI    = enc_x    + (size_t)LL*HD;  // L*3H
  float* enc_outs = encGI    + (size_t)LL*HD3; // L*H
  float* dec_e    = enc_outs + (size_t)LL*HD;  // L*H
  float* attn_e   = dec_e    + (size_t)LL*HD;  // L*L
  float* H2buf    = attn_e   + (size_t)LL*LL;  // L*H
  float* h0       = H2buf    + (size_t)LL*HD;  // H
  float* scores   = h0       + HD;             // L
  float* ctx      = scores   + LL;             // H
  float* cbuf     = ctx      + HD;             // H
  float* nll      = cbuf     + HD;             // L
  float* logits   = nll      + LL;             // L*VOUT

  // 1) Gather embeddings, zero h0.
  init_gather<<<LL, 256, 0, stream>>>(input_ids, target_ids, enc_emb, dec_emb,
                                      enc_x, dec_e, h0);

  // 2) Batched input-gate GEMM for encoder (WMMA f32):
  //    encGI[t, :] = enc_Wi @ enc_x[t] + enc_bi
  wmma_gemm_nt<<<dim3(HD3 / 256, LL / 16), 128, 0, stream>>>(
      enc_x, HD, enc_Wi, HD, enc_bi, encGI, HD3, HD);

  // 3) Batched attention e-half (WMMA f32):
  //    attn_e[t, l] = attn_W[l, :H] . dec_e[t] + attn_b[l]
  wmma_gemm_nt<<<dim3(LL / 256, LL / 16), 128, 0, stream>>>(
      dec_e, HD, attn_W, 2 * HD, attn_b, attn_e, LL, HD);

  // 4) Encoder recurrence (sequential; Wh stays L2-resident).
  for (int t = 0; t < LL; ++t) {
    const float* hp = (t == 0) ? h0 : (enc_outs + (size_t)(t - 1) * HD);
    gru_step_pre<<<HD / 8, 256, 0, stream>>>(enc_Wh, enc_bh,
                                             encGI + (size_t)t * HD3,
                                             hp, enc_outs + (size_t)t * HD);
  }

  // 5) Decoder recurrence with attention (sequential).
  for (int t = 0; t < LL; ++t) {
    const float* hp = (t == 0) ? (enc_outs + (size_t)(LL - 1) * HD)
                               : (H2buf + (size_t)(t - 1) * HD);
    const float* e_t = dec_e + (size_t)t * HD;
    attn_score<<<LL / 8, 256, 0, stream>>>(attn_W, attn_e + (size_t)t * LL, hp, scores);
    softmax_ctx<<<1, LL, 0, stream>>>(scores, enc_outs, ctx);
    combine<<<HD / 8, 256, 0, stream>>>(comb_W, comb_b, e_t, ctx, cbuf);
    gru_step_full<<<HD / 8, 256, 0, stream>>>(dec_Wi, dec_Wh, dec_bi, dec_bh,
                                              cbuf, hp, H2buf + (size_t)t * HD);
  }

  // 6) Output projection as ONE GEMM (WMMA f32) — out_W read once (131MB)
  //    instead of 256x: logits[t, v] = out_W[v] . H2[t] + out_b[v]
  wmma_gemm_nt<<<dim3(VOUT / 256, LL / 16), 128, 0, stream>>>(
      H2buf, HD, out_W, HD, out_b, logits, VOUT, HD);

  // 7) Per-step log-softmax NLL, then total.
  row_nll<<<LL, 256, 0, stream>>>(logits, target_ids, nll);
  final_sum<<<1, LL, 0, stream>>>(nll, out);
}